// k1_GNN_sep_7842610283373
// MI455X (gfx1250) — compile-verified
//
#include <hip/hip_runtime.h>
#include <hip/hip_bf16.h>
#include <math.h>

typedef __attribute__((ext_vector_type(2))) float v2f;
typedef __attribute__((ext_vector_type(8))) float v8f;

#define N_NODES  20000
#define N_EDGES  80000
#define N_GRAPHS 1000
#define F_TOT    16
#define EAD      5
#define CSD      5
#define FC_IN    75
#define GS       76   // 75 pooled features + 1 count slot

static inline int cdiv(int a, int b) { return (a + b - 1) / b; }

__device__ __forceinline__ float elu1(float x) {
  return x > 0.f ? x : (expf(x) - 1.f);
}

__device__ __forceinline__ v8f wmma_f32_k4(v2f a, v2f b, v8f c) {
  return __builtin_amdgcn_wmma_f32_16x16x4_f32(false, a, false, b, (short)0, c,
                                               false, false);
}

// h0 = x[:, :CSD]
__global__ void k_h0(const float* __restrict__ x, float* __restrict__ h0) {
  int idx = blockIdx.x * blockDim.x + threadIdx.x;
  if (idx >= N_NODES * CSD) return;
  int n = idx / CSD, i = idx - n * CSD;
  h0[idx] = x[(size_t)n * F_TOT + i];
}

// pre[n,o] = bias[o] + sum_i h[n,i] * root[i,o]
template <int MI, int MO>
__global__ void k_root(const float* __restrict__ h, const float* __restrict__ root,
                       const float* __restrict__ bias, float* __restrict__ pre) {
  int idx = blockIdx.x * blockDim.x + threadIdx.x;
  if (idx >= N_NODES * MO) return;
  int n = idx / MO, o = idx - n * MO;
  float s = bias[o];
  const float* hr = h + (size_t)n * MI;
  for (int i = 0; i < MI; ++i) s += hr[i] * root[i * MO + o];
  pre[idx] = s;
}

// Fused per-edge message as bilinear GEMM:
//   msg[e,o] = sum_{k,i} hid[e,k] * h_src[e,i] * w2[k, i*MO+o]  (+ h_src @ b2)
// A(16e x MI*128)[e, i*128+k] = hid[e,k]*h_src[e,i] built on the fly in LDS/regs,
// B(MI*128 x MO)[j,o] = w2[k, i*MO+o]. The b2 term is folded in as ceil(MI/4)
// appended K-steps (A=h_src, B=b2). Block = 64 threads = 2 waves over 64 edges;
// each wave takes half the output columns and all four 16-edge M-tiles, so each
// B fragment fetched from L2 serves 64 edges. Scatter via global_atomic_add_f32.
template <int MI, int MO>
__global__ __launch_bounds__(64) void k_msg(
    const float* __restrict__ h,      // N x MI
    const int*   __restrict__ eidx,   // 2 x E (row0 = src, row1 = dst)
    const float* __restrict__ eattr,  // E x 5
    const float* __restrict__ w1,     // 5 x 128
    const float* __restrict__ b1,     // 128
    const float* __restrict__ w2,     // 128 x (MI*MO)
    const float* __restrict__ b2,     // MI*MO
    float*       __restrict__ pre)    // N x MO (accumulated atomically)
{
  constexpr int NB   = MO / 16;       // 16-wide N-blocks total
  constexpr int NBH  = NB / 2;        // N-blocks per wave
  constexpr int MIP  = MI + 1;        // padded hsrc stride (bank-conflict free)
  constexpr int HS   = 129;           // padded hid stride
  constexpr int K4   = MI * 32;       // main K=4 steps (K_total = MI*128)
  constexpr int KE4  = (MI + 3) / 4;  // appended K=4 steps for the b2 term
  constexpr int MIMO = MI * MO;
  static_assert(NB % 2 == 0, "MO must be a multiple of 32");

  __shared__ float s_hid[64 * HS];    // hid tile, 64 edges x 128
  __shared__ float s_hsrc[64 * MIP];  // gathered h[src] rows
  __shared__ float s_w1[EAD * 128];
  __shared__ float s_b1[128];
  __shared__ float s_ea[64 * EAD];
  __shared__ int   s_src[64];
  __shared__ int   s_dst[64];

  const int tid  = threadIdx.x;
  const int base = blockIdx.x * 64;

  for (int idx = tid; idx < EAD * 128; idx += 64) s_w1[idx] = w1[idx];
  for (int idx = tid; idx < 128; idx += 64)       s_b1[idx] = b1[idx];
  {
    int e = base + tid;
    s_src[tid] = (e < N_EDGES) ? eidx[e] : 0;
    s_dst[tid] = (e < N_EDGES) ? eidx[N_EDGES + e] : 0;
  }
  for (int idx = tid; idx < 64 * EAD; idx += 64) {
    int el = idx / EAD, a = idx - el * EAD;
    int e = base + el;
    s_ea[idx] = (e < N_EDGES) ? eattr[(size_t)e * EAD + a] : 0.f;
  }
  __syncthreads();

  // gather h[src] rows (zeroed for padding edges -> their A rows are 0)
  for (int idx = tid; idx < 64 * MI; idx += 64) {
    int el = idx / MI, i = idx - el * MI;
    int e = base + el;
    float v = 0.f;
    if (e < N_EDGES) v = h[(size_t)s_src[el] * MI + i];
    s_hsrc[el * MIP + i] = v;
  }
  // hid tile: relu(edge_attr @ w1 + b1), recomputed in LDS (never hits HBM)
  for (int idx = tid; idx < 64 * 128; idx += 64) {
    int el = idx >> 7, k = idx & 127;
    float s = s_b1[k];
#pragma unroll
    for (int a = 0; a < EAD; ++a) s += s_ea[el * EAD + a] * s_w1[a * 128 + k];
    s_hid[el * HS + k] = fmaxf(s, 0.f);
  }
  __syncthreads();

  const int lane = tid & 31;
  const int wv   = tid >> 5;              // wave id (0..1): N-column half
  const int hh   = lane >> 4;             // lane half: K pair {0,1} vs {2,3}
  const int n16  = lane & 15;
  const int ob   = wv * (NBH * 16) + n16; // this wave's base output column

  v8f acc[4][NBH];
#pragma unroll
  for (int t = 0; t < 4; ++t)
#pragma unroll
    for (int nb = 0; nb < NBH; ++nb)
      acc[t][nb] = (v8f){0.f, 0.f, 0.f, 0.f, 0.f, 0.f, 0.f, 0.f};

  // main bilinear GEMM loop
  for (int ks = 0; ks < K4; ++ks) {
    const int j0 = ks * 4;
    const int i  = j0 >> 7;               // node-feature index (constant per step)
    const int kb = (j0 & 127) + 2 * hh;   // hid index for this lane's K slots
    v2f a[4];
#pragma unroll
    for (int t = 0; t < 4; ++t) {
      const int el = t * 16 + n16;
      const float hv = s_hsrc[el * MIP + i];
      a[t].x = s_hid[el * HS + kb]     * hv;
      a[t].y = s_hid[el * HS + kb + 1] * hv;
    }
    const float* brow = w2 + (size_t)kb * MIMO + (size_t)i * MO + ob;
#pragma unroll
    for (int nb = 0; nb < NBH; ++nb) {
      v2f b;
      b.x = brow[nb * 16];
      b.y = brow[nb * 16 + MIMO];
#pragma unroll
      for (int t = 0; t < 4; ++t)
        acc[t][nb] = wmma_f32_k4(a[t], b, acc[t][nb]);
    }
  }

  // folded-in b2 term: msg += h_src @ reshape(b2, MI, MO)
  for (int ks = 0; ks < KE4; ++ks) {
    const int jx = ks * 4 + 2 * hh;
    const int jy = jx + 1;
    const int ibx = (jx < MI) ? jx : (MI - 1);   // clamp to stay in bounds
    const int iby = (jy < MI) ? jy : (MI - 1);
    v2f a[4];
#pragma unroll
    for (int t = 0; t < 4; ++t) {
      const int el = t * 16 + n16;
      a[t].x = (jx < MI) ? s_hsrc[el * MIP + jx] : 0.f;
      a[t].y = (jy < MI) ? s_hsrc[el * MIP + jy] : 0.f;
    }
#pragma unroll
    for (int nb = 0; nb < NBH; ++nb) {
      v2f b;
      b.x = b2[ibx * MO + ob + nb * 16];
      b.y = b2[iby * MO + ob + nb * 16];
#pragma unroll
      for (int t = 0; t < 4; ++t)
        acc[t][nb] = wmma_f32_k4(a[t], b, acc[t][nb]);
    }
  }

  // C/D layout: reg r holds M = r + 8*half, N = lane&15. Scatter-add to pre[dst].
#pragma unroll
  for (int t = 0; t < 4; ++t) {
#pragma unroll
    for (int r = 0; r < 8; ++r) {
      const int el = t * 16 + r + 8 * hh;
      const int e  = base + el;
      if (e < N_EDGES) {
        const int dst = s_dst[el];
#pragma unroll
        for (int nb = 0; nb < NBH; ++nb)
          atomicAdd(&pre[(size_t)dst * MO + ob + nb * 16], acc[t][nb][r]);
      }
    }
  }
}

__global__ void k_elu(float* __restrict__ buf, int count) {
  int idx = blockIdx.x * blockDim.x + threadIdx.x;
  if (idx >= count) return;
  buf[idx] = elu1(buf[idx]);
}

__global__ void k_zero(float* __restrict__ buf, int count) {
  int idx = blockIdx.x * blockDim.x + threadIdx.x;
  if (idx < count) buf[idx] = 0.f;
}

// segment sums: gsum[g, 0:64]=sum h, [64:75]=sum x_cont, [75]=count
__global__ void k_seg(const float* __restrict__ h, const float* __restrict__ x,
                      const int* __restrict__ batch, float* __restrict__ gsum) {
  int idx = blockIdx.x * blockDim.x + threadIdx.x;
  if (idx >= N_NODES * GS) return;
  int n = idx / GS, c = idx - n * GS;
  int g = batch[n];
  float v;
  if (c < 64)      v = h[(size_t)n * 64 + c];
  else if (c < 75) v = x[(size_t)n * F_TOT + CSD + (c - 64)];
  else             v = 1.f;
  atomicAdd(&gsum[(size_t)g * GS + c], v);
}

// per-graph mean + 75->32->16->1 MLP (one thread per graph)
__global__ void k_mlp(const float* __restrict__ gsum,
                      const float* __restrict__ fc1_w, const float* __restrict__ fc1_b,
                      const float* __restrict__ fc2_w, const float* __restrict__ fc2_b,
                      const float* __restrict__ fc3_w, const float* __restrict__ fc3_b,
                      float* __restrict__ out) {
  int g = blockIdx.x * blockDim.x + threadIdx.x;
  if (g >= N_GRAPHS) return;
  const float inv = 1.f / fmaxf(gsum[(size_t)g * GS + 75], 1.f);
  float mean[FC_IN];
  for (int c = 0; c < FC_IN; ++c) mean[c] = gsum[(size_t)g * GS + c] * inv;
  float a1[32];
  for (int o = 0; o < 32; ++o) {
    float s = fc1_b[o];
    for (int c = 0; c < FC_IN; ++c) s += mean[c] * fc1_w[c * 32 + o];
    a1[o] = elu1(s);
  }
  float a2[16];
  for (int o = 0; o < 16; ++o) {
    float s = fc2_b[o];
    for (int c = 0; c < 32; ++c) s += a1[c] * fc2_w[c * 16 + o];
    a2[o] = elu1(s);
  }
  float s = fc3_b[0];
  for (int c = 0; c < 16; ++c) s += a2[c] * fc3_w[c];
  out[g] = s;
}

extern "C" void kernel_launch(void* const* d_in, const int* in_sizes, int n_in,
                              void* d_out, int out_size, void* d_ws, size_t ws_size,
                              hipStream_t stream) {
  const float* x     = (const float*)d_in[0];
  const int*   eidx  = (const int*)d_in[1];
  const float* eattr = (const float*)d_in[2];
  const int*   batch = (const int*)d_in[3];
  const float* c_w1[3]   = {(const float*)d_in[4],  (const float*)d_in[10], (const float*)d_in[16]};
  const float* c_b1[3]   = {(const float*)d_in[5],  (const float*)d_in[11], (const float*)d_in[17]};
  const float* c_w2[3]   = {(const float*)d_in[6],  (const float*)d_in[12], (const float*)d_in[18]};
  const float* c_b2[3]   = {(const float*)d_in[7],  (const float*)d_in[13], (const float*)d_in[19]};
  const float* c_root[3] = {(const float*)d_in[8],  (const float*)d_in[14], (const float*)d_in[20]};
  const float* c_bias[3] = {(const float*)d_in[9],  (const float*)d_in[15], (const float*)d_in[21]};
  const float* fc1_w = (const float*)d_in[22];
  const float* fc1_b = (const float*)d_in[23];
  const float* fc2_w = (const float*)d_in[24];
  const float* fc2_b = (const float*)d_in[25];
  const float* fc3_w = (const float*)d_in[26];
  const float* fc3_b = (const float*)d_in[27];

  float* hbuf0 = (float*)d_ws;                 // N x 64 (max)
  float* hbuf1 = hbuf0 + (size_t)N_NODES * 64; // N x 64
  float* gsum  = hbuf1 + (size_t)N_NODES * 64; // G x GS

  const int msg_blocks = cdiv(N_EDGES, 64);

  // h0 = x[:, :5]
  k_h0<<<cdiv(N_NODES * CSD, 256), 256, 0, stream>>>(x, hbuf0);

  // ---- layer 1: (5 -> 32), h in hbuf0, pre/out in hbuf1 ----
  k_root<5, 32><<<cdiv(N_NODES * 32, 256), 256, 0, stream>>>(hbuf0, c_root[0], c_bias[0], hbuf1);
  k_msg<5, 32><<<msg_blocks, 64, 0, stream>>>(hbuf0, eidx, eattr, c_w1[0], c_b1[0],
                                              c_w2[0], c_b2[0], hbuf1);
  k_elu<<<cdiv(N_NODES * 32, 256), 256, 0, stream>>>(hbuf1, N_NODES * 32);

  // ---- layer 2: (32 -> 64), h in hbuf1, pre/out in hbuf0 ----
  k_root<32, 64><<<cdiv(N_NODES * 64, 256), 256, 0, stream>>>(hbuf1, c_root[1], c_bias[1], hbuf0);
  k_msg<32, 64><<<msg_blocks, 64, 0, stream>>>(hbuf1, eidx, eattr, c_w1[1], c_b1[1],
                                               c_w2[1], c_b2[1], hbuf0);
  k_elu<<<cdiv(N_NODES * 64, 256), 256, 0, stream>>>(hbuf0, N_NODES * 64);

  // ---- layer 3: (64 -> 64), h in hbuf0, pre/out in hbuf1 ----
  k_root<64, 64><<<cdiv(N_NODES * 64, 256), 256, 0, stream>>>(hbuf0, c_root[2], c_bias[2], hbuf1);
  k_msg<64, 64><<<msg_blocks, 64, 0, stream>>>(hbuf0, eidx, eattr, c_w1[2], c_b1[2],
                                               c_w2[2], c_b2[2], hbuf1);
  k_elu<<<cdiv(N_NODES * 64, 256), 256, 0, stream>>>(hbuf1, N_NODES * 64);

  // ---- readout ----
  k_zero<<<cdiv(N_GRAPHS * GS, 256), 256, 0, stream>>>(gsum, N_GRAPHS * GS);
  k_seg<<<cdiv(N_NODES * GS, 256), 256, 0, stream>>>(hbuf1, x, batch, gsum);
  k_mlp<<<cdiv(N_GRAPHS, 128), 128, 0, stream>>>(gsum, fc1_w, fc1_b, fc2_w, fc2_b,
                                                 fc3_w, fc3_b, (float*)d_out);
}